// SimpleGATEncoder_75161927680559
// MI455X (gfx1250) — compile-verified
//
#include <hip/hip_runtime.h>
#include <math.h>

#define D 128
#define H 4
#define C 32
#define LAYERS 3
#define SLOPE 0.2f
#define BN_EPS 1e-5f

typedef __attribute__((ext_vector_type(2))) float v2f;
typedef __attribute__((ext_vector_type(8))) float v8f;
typedef __attribute__((ext_vector_type(4))) int   v4i;

typedef __attribute__((address_space(1))) v4i* as1_v4i_ptr;   // "__device__" AS
typedef __attribute__((address_space(3))) v4i* as3_v4i_ptr;   // "__shared__" AS

__device__ __forceinline__ float lrelu(float v) {
    return v > 0.0f ? v : SLOPE * v;
}

// float atomic max via integer punning (valid when init = -inf)
__device__ __forceinline__ void atomicMaxF(float* addr, float val) {
    if (val >= 0.0f) {
        atomicMax((int*)addr, __float_as_int(val));
    } else {
        atomicMin((unsigned int*)addr, __float_as_uint(val));
    }
}

// gfx1250 async global->LDS copy of 16 bytes (ASYNCcnt-tracked).
__device__ __forceinline__ void async_copy_b128(const float* gsrc, float* lds_dst) {
#if __has_builtin(__builtin_amdgcn_global_load_async_to_lds_b128)
    __builtin_amdgcn_global_load_async_to_lds_b128(
        (as1_v4i_ptr)gsrc, (as3_v4i_ptr)lds_dst, 0, 0);
#else
    asm volatile("global_load_async_to_lds_b128 %0, %1, off"
                 :: "v"((unsigned)(size_t)lds_dst), "v"(gsrc)
                 : "memory");
#endif
}

__device__ __forceinline__ void wait_asynccnt0() {
#if __has_builtin(__builtin_amdgcn_s_wait_asynccnt)
    __builtin_amdgcn_s_wait_asynccnt(0);
#else
    asm volatile("s_wait_asynccnt 0x0" ::: "memory");
#endif
}

// ---------------------------------------------------------------------------
// GEMM: Y[n x 128] = X[n x 128] @ W[128 x 128] (+ bias[128])
// Block = 256 threads = 8 waves = 16 rows x 128 cols.
// The 16x128 X tile is staged in LDS once per block via async load-to-LDS
// (8x reduction of global A traffic); each wave then runs 32 x
// V_WMMA_F32_16X16X4_F32 with A-frags from LDS and B-frags from global (L2).
// ---------------------------------------------------------------------------
__global__ __launch_bounds__(256) void gemm128_wmma(
        const float* __restrict__ X, const float* __restrict__ W,
        const float* __restrict__ bias, float* __restrict__ Y, int n) {
    __shared__ float sX[16 * D];              // 8 KB

    const int tid   = threadIdx.x;
    const int lane  = tid & 31;
    const int wave  = tid >> 5;
    const int r0    = blockIdx.x * 16;
    const int c0    = wave * 16;
    const int mrow  = lane & 15;
    const int khalf = (lane >> 4) << 1;       // 0 or 2

    // Cooperative async stage of X[r0 .. r0+15][0..127] -> sX (2 x b128/thread).
#pragma unroll
    for (int i = 0; i < 2; ++i) {
        int f4   = tid + i * 256;             // 16B chunk id, 512 total
        int fidx = f4 << 2;                   // float index in tile
        int row  = fidx >> 7;
        int col  = fidx & (D - 1);
        int grow = r0 + row;
        if (grow >= n) grow = n - 1;          // clamp (exact for N=50000)
        async_copy_b128(X + (size_t)grow * D + col, &sX[fidx]);
    }
    wait_asynccnt0();
    __syncthreads();

    const int bcol = c0 + mrow;
    const float* arow = &sX[mrow * D];

    v8f acc = {};
#pragma unroll
    for (int k = 0; k < D; k += 4) {
        v2f a, b;
        a.x = arow[k + khalf + 0];
        a.y = arow[k + khalf + 1];
        b.x = W[(size_t)(k + khalf + 0) * D + bcol];
        b.y = W[(size_t)(k + khalf + 1) * D + bcol];
        acc = __builtin_amdgcn_wmma_f32_16x16x4_f32(
            false, a, false, b, (short)0, acc, false, false);
    }

    const float badd = bias ? bias[bcol] : 0.0f;
    const int rbase = r0 + ((lane >> 4) << 3);  // lanes 16-31 hold M = r+8
#pragma unroll
    for (int r = 0; r < 8; ++r) {
        int row = rbase + r;
        if (row < n) Y[(size_t)row * D + bcol] = acc[r] + badd;
    }
}

// ---------------------------------------------------------------------------
// Build int32 src/dst lists with self-loops appended.
// ---------------------------------------------------------------------------
__global__ void prep_edges(const int* __restrict__ ei, int* __restrict__ src,
                           int* __restrict__ dst, int E, int n) {
    int t = blockIdx.x * blockDim.x + threadIdx.x;
    int ET = E + n;
    if (t >= ET) return;
    if (t < E) { src[t] = ei[t]; dst[t] = ei[E + t]; }
    else       { src[t] = t - E; dst[t] = t - E; }
}

// Per-layer re-init: segment max = -inf, denom = 0, BN stats = 0.
__global__ void init_layer(float* __restrict__ mx, float* __restrict__ den,
                           float* __restrict__ stats, int NH) {
    int t = blockIdx.x * blockDim.x + threadIdx.x;
    if (t < NH) { mx[t] = -__builtin_inff(); den[t] = 0.0f; }
    if (t < 4 * D) stats[t] = 0.0f;
}

// ---------------------------------------------------------------------------
// Edge pass 1: e[eh] = att_h . leakyrelu(xl[src] + xr[dst]); segment max.
// ---------------------------------------------------------------------------
__global__ __launch_bounds__(256) void edge_score(
        const float* __restrict__ XL, const float* __restrict__ XR,
        const float* __restrict__ att, const int* __restrict__ src,
        const int* __restrict__ dst, float* __restrict__ esc,
        float* __restrict__ mx, int ET) {
    int t = blockIdx.x * blockDim.x + threadIdx.x;
    if (t >= ET * H) return;
    int e = t >> 2, h = t & (H - 1);
    int s = src[e], d = dst[e];
    const float4* pl = (const float4*)(XL + (size_t)s * D + h * C);
    const float4* pr = (const float4*)(XR + (size_t)d * D + h * C);
    const float4* pa = (const float4*)(att + h * C);
    float acc = 0.0f;
#pragma unroll
    for (int i = 0; i < C / 4; ++i) {
        float4 a = pl[i], b = pr[i], w = pa[i];
        acc += lrelu(a.x + b.x) * w.x + lrelu(a.y + b.y) * w.y
             + lrelu(a.z + b.z) * w.z + lrelu(a.w + b.w) * w.w;
    }
    esc[t] = acc;
    atomicMaxF(&mx[d * H + h], acc);
}

// Edge pass 2: ex = exp(e - m[dst]); segment sum of ex.
__global__ __launch_bounds__(256) void edge_exp(
        const int* __restrict__ dst, float* __restrict__ esc,
        const float* __restrict__ mx, float* __restrict__ den, int ET) {
    int t = blockIdx.x * blockDim.x + threadIdx.x;
    if (t >= ET * H) return;
    int e = t >> 2, h = t & (H - 1);
    int d = dst[e];
    float ex = __expf(esc[t] - mx[d * H + h]);
    esc[t] = ex;
    unsafeAtomicAdd(&den[d * H + h], ex);
}

__global__ void invert(float* __restrict__ den, int NH) {
    int t = blockIdx.x * blockDim.x + threadIdx.x;
    if (t < NH) den[t] = 1.0f / den[t];   // safe: self-loops => den > 0
}

// Edge pass 3: AGG[dst] += alpha * xl[src]. 32 threads/edge, 4 ch each.
__global__ __launch_bounds__(256) void edge_agg(
        const float* __restrict__ XL, const int* __restrict__ src,
        const int* __restrict__ dst, const float* __restrict__ esc,
        const float* __restrict__ invden, float* __restrict__ AGG, int ET) {
    int t = blockIdx.x * blockDim.x + threadIdx.x;
    if (t >= ET * 32) return;
    int e = t >> 5, q = t & 31;
    int ch = q << 2, h = q >> 3;
    int s = src[e], d = dst[e];
    float alpha = esc[e * H + h] * invden[d * H + h];
    float4 v = *(const float4*)(XL + (size_t)s * D + ch);
    float* out = AGG + (size_t)d * D + ch;
    unsafeAtomicAdd(out + 0, alpha * v.x);
    unsafeAtomicAdd(out + 1, alpha * v.y);
    unsafeAtomicAdd(out + 2, alpha * v.z);
    unsafeAtomicAdd(out + 3, alpha * v.w);
}

// ---------------------------------------------------------------------------
// BatchNorm: per-channel sum/sumsq (thread = channel -> coalesced).
// ---------------------------------------------------------------------------
__global__ __launch_bounds__(128) void bn_stats(
        const float* __restrict__ A, float* __restrict__ stats, int n) {
    int ch = threadIdx.x;                 // 0..127
    int row0 = blockIdx.x * 64;
    int rowN = row0 + 64 < n ? row0 + 64 : n;
    float s = 0.0f, ss = 0.0f;
    for (int r = row0; r < rowN; ++r) {
        float v = A[(size_t)r * D + ch];
        s += v; ss += v * v;
    }
    unsafeAtomicAdd(&stats[ch], s);
    unsafeAtomicAdd(&stats[D + ch], ss);
}

__global__ void bn_final(const float* __restrict__ gamma,
                         const float* __restrict__ beta,
                         float* __restrict__ stats, int n) {
    int ch = threadIdx.x;                 // single block of 128
    float inv_n = 1.0f / (float)n;
    float mean = stats[ch] * inv_n;
    float var  = stats[D + ch] * inv_n - mean * mean;
    float sc   = rsqrtf(var + BN_EPS) * gamma[ch];
    stats[2 * D + ch] = sc;
    stats[3 * D + ch] = beta[ch] - mean * sc;
}

__global__ __launch_bounds__(256) void bn_apply(
        const float* __restrict__ A, const float* __restrict__ stats,
        float* __restrict__ Xn, int total) {
    int t = blockIdx.x * blockDim.x + threadIdx.x;
    if (t >= total) return;
    int ch = t & (D - 1);
    float y = A[t] * stats[2 * D + ch] + stats[3 * D + ch];
    Xn[t] = lrelu(y);
}

// ---------------------------------------------------------------------------
extern "C" void kernel_launch(void* const* d_in, const int* in_sizes, int n_in,
                              void* d_out, int out_size, void* d_ws, size_t ws_size,
                              hipStream_t stream) {
    const float* x     = (const float*)d_in[0];
    const int*   ei    = (const int*)  d_in[1];
    const float* Wl    = (const float*)d_in[2];
    const float* Wr    = (const float*)d_in[3];
    const float* att   = (const float*)d_in[4];
    const float* bias  = (const float*)d_in[5];
    const float* Wres  = (const float*)d_in[6];
    const float* gamma = (const float*)d_in[7];
    const float* beta  = (const float*)d_in[8];
    const float* Wout  = (const float*)d_in[9];
    const float* bout  = (const float*)d_in[10];

    const int n  = in_sizes[0] / D;
    const int E  = in_sizes[1] / 2;
    const int ET = E + n;

    float* f = (float*)d_ws;
    size_t off = 0;
    float* Xc    = f + off; off += (size_t)n * D;   // current node features
    float* XL    = f + off; off += (size_t)n * D;   // x @ Wl
    float* XR    = f + off; off += (size_t)n * D;   // x @ Wr
    float* AGG   = f + off; off += (size_t)n * D;   // residual + aggregated msgs
    float* ESC   = f + off; off += (size_t)ET * H;  // edge scores / exp weights
    float* MX    = f + off; off += (size_t)n * H;   // segment max
    float* DEN   = f + off; off += (size_t)n * H;   // segment sum -> inverse
    float* STATS = f + off; off += 4 * D;           // BN sum/sumsq/scale/shift
    int* SRC = (int*)(f + off); off += ET;
    int* DST = (int*)(f + off); off += ET;

    const int TB = 256;
    prep_edges<<<(ET + TB - 1) / TB, TB, 0, stream>>>(ei, SRC, DST, E, n);

    const float* xin = x;
    const int gemm_blocks = (n + 15) / 16;
    for (int l = 0; l < LAYERS; ++l) {
        const size_t wo = (size_t)l * D * D;
        gemm128_wmma<<<gemm_blocks, 256, 0, stream>>>(xin, Wl + wo, nullptr, XL, n);
        gemm128_wmma<<<gemm_blocks, 256, 0, stream>>>(xin, Wr + wo, nullptr, XR, n);
        // AGG seeded with residual + bias; messages scatter-added on top.
        gemm128_wmma<<<gemm_blocks, 256, 0, stream>>>(xin, Wres + wo, bias + (size_t)l * D, AGG, n);

        init_layer<<<(n * H + TB - 1) / TB, TB, 0, stream>>>(MX, DEN, STATS, n * H);
        edge_score<<<(ET * H + TB - 1) / TB, TB, 0, stream>>>(
            XL, XR, att + (size_t)l * H * C, SRC, DST, ESC, MX, ET);
        edge_exp<<<(ET * H + TB - 1) / TB, TB, 0, stream>>>(DST, ESC, MX, DEN, ET);
        invert<<<(n * H + TB - 1) / TB, TB, 0, stream>>>(DEN, n * H);
        edge_agg<<<(ET * 32 + TB - 1) / TB, TB, 0, stream>>>(
            XL, SRC, DST, ESC, DEN, AGG, ET);

        bn_stats<<<(n + 63) / 64, 128, 0, stream>>>(AGG, STATS, n);
        bn_final<<<1, 128, 0, stream>>>(gamma + (size_t)l * D, beta + (size_t)l * D, STATS, n);
        bn_apply<<<(n * D + TB - 1) / TB, TB, 0, stream>>>(AGG, STATS, Xc, n * D);
        xin = Xc;
    }

    gemm128_wmma<<<gemm_blocks, 256, 0, stream>>>(xin, Wout, bout, (float*)d_out, n);
}